// Yolo_59966333387037
// MI455X (gfx1250) — compile-verified
//
#include <hip/hip_runtime.h>
#include <cstdint>
#include <cstddef>

// ---- problem constants (match reference) ----
#define NCLS   80
#define A_     3
#define CH     85            // 5 + NCLS
#define H_     76
#define W_     76
#define HW     (H_ * W_)     // 5776
#define BATCH  16
#define IN_WH  608.0f

#define POS        64                        // spatial positions per tile
#define CHUNKS     ((HW + POS - 1) / POS)    // 91 (last tile: 16 valid positions)
#define TILE_ELEMS (CH * POS)                // 5440 floats = 21,760 B of LDS
#define THREADS    256

// ---------------- CDNA5-specific helpers ----------------

// Async DMA: global -> LDS, 4 bytes per lane, tracked by ASYNCcnt.
// vdst = LDS byte address VGPR, vaddr = 64-bit global address VGPR pair.
// Per-lane LDS addresses let the DMA perform the transpose for free.
__device__ __forceinline__ void async_load_b32(const float* gaddr, uint32_t lds_byte_off) {
    asm volatile("global_load_async_to_lds_b32 %0, %1, off"
                 :: "v"(lds_byte_off), "v"(gaddr)
                 : "memory");
}

__device__ __forceinline__ void wait_async0() {
#if __has_builtin(__builtin_amdgcn_s_wait_asynccnt)
    __builtin_amdgcn_s_wait_asynccnt(0);
#else
    asm volatile("s_wait_asynccnt 0x0" ::: "memory");
#endif
}

// sigmoid(x) = 0.5*tanh(x/2) + 0.5 -> single V_TANH_F32 trans op on CDNA5
__device__ __forceinline__ float sigmoid_f(float x) {
#if __has_builtin(__builtin_amdgcn_tanhf)
    return __builtin_fmaf(0.5f, __builtin_amdgcn_tanhf(0.5f * x), 0.5f);
#elif __has_builtin(__builtin_amdgcn_tanh_f32)
    return __builtin_fmaf(0.5f, __builtin_amdgcn_tanh_f32(0.5f * x), 0.5f);
#else
    float t, h = 0.5f * x;
    asm("v_tanh_f32 %0, %1" : "=v"(t) : "v"(h));
    return __builtin_fmaf(0.5f, t, 0.5f);
#endif
}

__device__ __forceinline__ float decode_one(float raw, int c, int p, int hw0,
                                            float aw_s, float ah_s,
                                            float inv_w, float inv_h) {
    if (c >= 4) return sigmoid_f(raw);           // conf + class probs (81/85 of elems)
    if (c == 2) return __expf(raw) * aw_s;       // bw = exp(pw)*aw/IN_W
    if (c == 3) return __expf(raw) * ah_s;       // bh = exp(ph)*ah/IN_H
    const int hw = hw0 + p;
    const int gy = hw / W_;
    if (c == 0) {                                // bx = (sigmoid+gx)/W
        const int gx = hw - gy * W_;
        return (sigmoid_f(raw) + (float)gx) * inv_w;
    }
    return (sigmoid_f(raw) + (float)gy) * inv_h; // by = (sigmoid+gy)/H
}

// ---------------- kernel ----------------
// One block decodes an 85-channel x 64-position tile of one (n, a) map.
// Phase 1: async DMA scatters the channel-major tile into LDS ALREADY in the
//          channel-last output layout [p][c] (reads coalesced 128B/wave; LDS
//          writes stride 85 dwords -> 85 mod 64 = 21 -> conflict-free).
// Phase 2: contiguous 16B-aligned ds_load_b128 from LDS, decode math,
//          coalesced global_store_b128 of the contiguous output tile.
__global__ __launch_bounds__(THREADS)
void yolo_decode_kernel(const float* __restrict__ x,
                        const float* __restrict__ anchors,
                        float* __restrict__ out) {
    __shared__ __align__(16) float lds[TILE_ELEMS];   // flat [p][c], stride 85

    const int b     = blockIdx.x;
    const int chunk = b % CHUNKS;
    const int na    = b / CHUNKS;            // n*A + a
    const int a     = na % A_;
    const int hw0   = chunk * POS;
    const int validP = (HW - hw0) < POS ? (HW - hw0) : POS;

    const float aw_s  = anchors[2 * a + 0] * (1.0f / IN_WH);
    const float ah_s  = anchors[2 * a + 1] * (1.0f / IN_WH);
    const float inv_w = 1.0f / (float)W_;
    const float inv_h = 1.0f / (float)H_;

    const int tid = threadIdx.x;

    // ---- Phase 1: global (channel-major) -> LDS (channel-last) via async DMA
    const float* xbase = x + (size_t)na * CH * HW + hw0;
    for (int t = tid; t < TILE_ELEMS; t += THREADS) {
        const int c  = t >> 6;                          // t / POS
        const int p  = t & (POS - 1);                   // t % POS (lane-consecutive)
        const int pg = p < validP ? p : (validP - 1);   // clamp: stay in-bounds
        const uint32_t loff = (uint32_t)(size_t)&lds[p * CH + c];  // transpose here
        async_load_b32(xbase + (size_t)c * HW + pg, loff);
    }
    wait_async0();
    __syncthreads();

    // ---- Phase 2: vector LDS read -> decode -> vector global store ----
    float* obase = out + ((size_t)na * HW + hw0) * CH;  // 16B-aligned tile base
    const int nOut = validP * CH;                       // 5440 or 1360 (both %4==0)

    for (int f4 = tid * 4; f4 + 3 < nOut; f4 += THREADS * 4) {
        const float4 raw = *(const float4*)&lds[f4];    // ds_load_b128, aligned
        int p = f4 / CH;
        int c = f4 - p * CH;
        float v[4];
        const float rr[4] = { raw.x, raw.y, raw.z, raw.w };
#pragma unroll
        for (int j = 0; j < 4; ++j) {
            v[j] = decode_one(rr[j], c, p, hw0, aw_s, ah_s, inv_w, inv_h);
            if (++c == CH) { c = 0; ++p; }
        }
        *(float4*)(obase + f4) = make_float4(v[0], v[1], v[2], v[3]);
    }

    // scalar tail (never taken with these dims; kept for safety)
    for (int f = (nOut & ~3) + tid; f < nOut; f += THREADS) {
        const int p = f / CH;
        const int c = f - p * CH;
        obase[f] = decode_one(lds[f], c, p, hw0, aw_s, ah_s, inv_w, inv_h);
    }
}

// ---------------- host launch ----------------
extern "C" void kernel_launch(void* const* d_in, const int* in_sizes, int n_in,
                              void* d_out, int out_size, void* d_ws, size_t ws_size,
                              hipStream_t stream) {
    const float* x       = (const float*)d_in[0];
    const float* anchors = (const float*)d_in[1];
    float*       out     = (float*)d_out;

    const int grid = BATCH * A_ * CHUNKS;     // 16 * 3 * 91 = 4368 blocks
    yolo_decode_kernel<<<grid, THREADS, 0, stream>>>(x, anchors, out);
}